// Laplacian_transformation_grid_add_1786706395328
// MI455X (gfx1250) — compile-verified
//
#include <hip/hip_runtime.h>
#include <stdint.h>

#define NIMG 8
#define NCH  3
#define S0 512
#define S1 256
#define S2 128
#define S3 64

#if __has_builtin(__builtin_amdgcn_global_load_async_to_lds_b128)
#define HAVE_ASYNC_LDS 1
typedef __attribute__((ext_vector_type(4))) int v4i_t;
typedef __attribute__((address_space(1))) v4i_t* gptr_v4i;
typedef __attribute__((address_space(3))) v4i_t* lptr_v4i;
#endif

// ---------------------------------------------------------------------------
// Bilinear helpers (align_corners=True resize)
// ---------------------------------------------------------------------------
__device__ __forceinline__ void bl_ac(float coord, int Smax, int& i0, int& i1, float& w) {
  float f = floorf(coord);
  i0 = (int)f;
  w  = coord - f;
  i1 = min(i0 + 1, Smax - 1);
  if (i0 > Smax - 1) i0 = Smax - 1;   // fp safety at the top edge
}

// dst[nc, oy, ox] = bilinear_ac(src at Hin x Hin -> Hout x Hout)
// Compile-time sizes: index decode becomes shifts, scale a literal.
template <int Hin, int Hout>
__global__ __launch_bounds__(256) void resize_ac_kernel(const float* __restrict__ src,
                                                        float* __restrict__ dst) {
  int idx = blockIdx.x * blockDim.x + threadIdx.x;
  constexpr int total = NIMG * NCH * Hout * Hout;
  if (idx >= total) return;
  int ox = idx & (Hout - 1);
  int oy = (idx / Hout) & (Hout - 1);
  int nc = idx / (Hout * Hout);
  constexpr float scale = (float)(Hin - 1) / (float)(Hout - 1);
  int y0, y1, x0, x1; float wy, wx;
  bl_ac(oy * scale, Hin, y0, y1, wy);
  bl_ac(ox * scale, Hin, x0, x1, wx);
  const float* s = src + (size_t)nc * (Hin * Hin);
  float r0 = s[y0 * Hin + x0] * (1.f - wy) + s[y1 * Hin + x0] * wy;
  float r1 = s[y0 * Hin + x1] * (1.f - wy) + s[y1 * Hin + x1] * wy;
  dst[idx] = r0 * (1.f - wx) + r1 * wx;
}

// out[nc, oy, ox] = base[nc, oy, ox] - bilinear_ac_upsample(coarse: Hc -> Hb)
template <int Hb, int Hc>
__global__ __launch_bounds__(256) void lap_kernel(const float* __restrict__ base,
                                                  const float* __restrict__ coarse,
                                                  float* __restrict__ out) {
  int idx = blockIdx.x * blockDim.x + threadIdx.x;
  constexpr int total = NIMG * NCH * Hb * Hb;
  if (idx >= total) return;
  int ox = idx & (Hb - 1);
  int oy = (idx / Hb) & (Hb - 1);
  int nc = idx / (Hb * Hb);
  constexpr float scale = (float)(Hc - 1) / (float)(Hb - 1);
  int y0, y1, x0, x1; float wy, wx;
  bl_ac(oy * scale, Hc, y0, y1, wy);
  bl_ac(ox * scale, Hc, x0, x1, wx);
  const float* s = coarse + (size_t)nc * (Hc * Hc);
  float r0 = s[y0 * Hc + x0] * (1.f - wy) + s[y1 * Hc + x0] * wy;
  float r1 = s[y0 * Hc + x1] * (1.f - wy) + s[y1 * Hc + x1] * wy;
  out[idx] = base[idx] - (r0 * (1.f - wx) + r1 * wx);
}

// ---------------------------------------------------------------------------
// grid_sample (bilinear, zeros padding, align_corners=False) accumulate.
// S is a compile-time constant at every call site after inlining.
// ---------------------------------------------------------------------------
__device__ __forceinline__ void sample_add(const float* __restrict__ base, int S,
                                           float gx, float gy, float acc[NCH]) {
  float ix = ((gx + 1.f) * (float)S - 1.f) * 0.5f;
  float iy = ((gy + 1.f) * (float)S - 1.f) * 0.5f;
  float fx = floorf(ix), fy = floorf(iy);
  int x0 = (int)fx, y0 = (int)fy;
  float wx = ix - fx, wy = iy - fy;
  int x1 = x0 + 1, y1 = y0 + 1;
  bool bx0 = ((unsigned)x0 < (unsigned)S);
  bool bx1 = ((unsigned)x1 < (unsigned)S);
  bool by0 = ((unsigned)y0 < (unsigned)S);
  bool by1 = ((unsigned)y1 < (unsigned)S);
  float w00 = (1.f - wy) * (1.f - wx);
  float w01 = (1.f - wy) * wx;
  float w10 = wy * (1.f - wx);
  float w11 = wy * wx;
  int cs  = S * S;
  int o00 = y0 * S + x0, o01 = y0 * S + x1;
  int o10 = y1 * S + x0, o11 = y1 * S + x1;
#pragma unroll
  for (int c = 0; c < NCH; ++c) {
    const float* b = base + c * cs;
    float v = 0.f;
    if (by0 && bx0) v += w00 * b[o00];
    if (by0 && bx1) v += w01 * b[o01];
    if (by1 && bx0) v += w10 * b[o10];
    if (by1 && bx1) v += w11 * b[o11];
    acc[c] += v;
  }
}

__global__ __launch_bounds__(256) void gridsample_sum_kernel(
    const float* __restrict__ lp0, const float* __restrict__ lp1,
    const float* __restrict__ lp2, const float* __restrict__ lp3,
    const float* __restrict__ theta, float* __restrict__ out) {
  __shared__ float s_lp3[NCH * S3 * S3];  // 48 KB: whole LP3 for this image

  const int tid = threadIdx.x;
  const int bx  = blockIdx.x;
  const int n   = bx >> 10;        // 1024 tiles (16x64 of 32x8) per image
  const int t   = bx & 1023;
  const int ti  = t >> 4;          // 0..63 tile rows (8 px tall)
  const int tj  = t & 15;          // 0..15 tile cols (32 px wide)

  // ---- stage LP3 of image n into LDS (CDNA5 async global->LDS, ASYNCcnt)
  const float* g3 = lp3 + (size_t)n * (NCH * S3 * S3);
#ifdef HAVE_ASYNC_LDS
  for (int k = tid * 4; k < NCH * S3 * S3; k += 256 * 4) {
    __builtin_amdgcn_global_load_async_to_lds_b128(
        (gptr_v4i)(g3 + k),           // global src (per-lane addr)
        (lptr_v4i)(&s_lp3[k]),        // LDS dst (per-lane addr)
        0, 0);
  }
#if __has_builtin(__builtin_amdgcn_s_wait_asynccnt)
  __builtin_amdgcn_s_wait_asynccnt(0);
#else
  asm volatile("s_wait_asynccnt 0" ::: "memory");
#endif
#else
  {
    const float4* g4 = (const float4*)g3;
    float4* s4 = (float4*)s_lp3;
    for (int k = tid; k < (NCH * S3 * S3) / 4; k += 256) s4[k] = g4[k];
  }
#endif
  __syncthreads();

  // ---- affine grid point (align_corners=True linspace over 512)
  const float* th = theta + n * 6;
  float t00 = th[0], t01 = th[1], t02 = th[2];
  float t10 = th[3], t11 = th[4], t12 = th[5];

  int i = (ti << 3) + (tid >> 5);   // row:   64 tiles * 8
  int j = (tj << 5) + (tid & 31);   // col:   16 tiles * 32 (wave-contiguous)
  const float inv = 2.0f / (float)(S0 - 1);
  float X = -1.f + (float)j * inv;
  float Y = -1.f + (float)i * inv;
  float gx = t00 * X + t01 * Y + t02;
  float gy = t10 * X + t11 * Y + t12;

  float acc[NCH] = {0.f, 0.f, 0.f};
  sample_add(lp0 + (size_t)n * (NCH * S0 * S0), S0, gx, gy, acc);
  sample_add(lp1 + (size_t)n * (NCH * S1 * S1), S1, gx, gy, acc);
  sample_add(lp2 + (size_t)n * (NCH * S2 * S2), S2, gx, gy, acc);
  sample_add(s_lp3, S3, gx, gy, acc);   // LDS-resident level

#pragma unroll
  for (int c = 0; c < NCH; ++c)
    out[(((size_t)n * NCH + c) * S0 + i) * S0 + j] = acc[c];
}

// ---------------------------------------------------------------------------
extern "C" void kernel_launch(void* const* d_in, const int* in_sizes, int n_in,
                              void* d_out, int out_size, void* d_ws, size_t ws_size,
                              hipStream_t stream) {
  (void)in_sizes; (void)n_in; (void)out_size; (void)ws_size;
  const float* image = (const float*)d_in[0];
  const float* pmtrx = (const float*)d_in[1];
  float* ws = (float*)d_ws;

  const size_t szL0 = (size_t)NIMG * NCH * S0 * S0;  // 6,291,456
  const size_t szL1 = (size_t)NIMG * NCH * S1 * S1;  // 1,572,864
  const size_t szL2 = (size_t)NIMG * NCH * S2 * S2;  //   393,216
  const size_t szL3 = (size_t)NIMG * NCH * S3 * S3;  //    98,304

  float* gp1 = ws;
  float* gp2 = gp1 + szL1;
  float* gp3 = gp2 + szL2;          // gp3 == LP3
  float* lp0 = gp3 + szL3;
  float* lp1 = lp0 + szL0;
  float* lp2 = lp1 + szL1;

  const int B = 256;
  // Gaussian pyramid
  resize_ac_kernel<S0, S1><<<(int)((szL1 + B - 1) / B), B, 0, stream>>>(image, gp1);
  resize_ac_kernel<S1, S2><<<(int)((szL2 + B - 1) / B), B, 0, stream>>>(gp1, gp2);
  resize_ac_kernel<S2, S3><<<(int)((szL3 + B - 1) / B), B, 0, stream>>>(gp2, gp3);
  // Laplacian pyramid
  lap_kernel<S0, S1><<<(int)((szL0 + B - 1) / B), B, 0, stream>>>(image, gp1, lp0);
  lap_kernel<S1, S2><<<(int)((szL1 + B - 1) / B), B, 0, stream>>>(gp1, gp2, lp1);
  lap_kernel<S2, S3><<<(int)((szL2 + B - 1) / B), B, 0, stream>>>(gp2, gp3, lp2);
  // Fused grid generation + 4-level grid_sample + sum
  gridsample_sum_kernel<<<NIMG * 1024, B, 0, stream>>>(lp0, lp1, lp2, gp3, pmtrx,
                                                       (float*)d_out);
}